// FourierAttention1d_44169443672696
// MI455X (gfx1250) — compile-verified
//
#include <hip/hip_runtime.h>

// ---------------------------------------------------------------------------
// FourierAttention1d for MI455X (gfx1250, wave32, WMMA f32 16x16x4)
//
//   B=4, N=8192, CIN=64, H=8, D=64, hd=512
//   Q = LN(v @ Wq^T + bq), K = LN(v @ Wk^T + bk), V = v @ Wv^T + bv
//   KV[b,h] = K^T V   (64x64),  M[b,h] = KV[b,h] @ o_w_h^T  (fold out-proj)
//   out = (sum_h Q_h @ M_h + o_b) / N
// ---------------------------------------------------------------------------

typedef __attribute__((ext_vector_type(2))) float v2f;
typedef __attribute__((ext_vector_type(8))) float v8f;

#define B_   4
#define N_   8192
#define CIN_ 64
#define H_   8
#define D_   64
#define XS   68      // LDS stride (floats): 272B = multiple of 16B, banks offset by 4
#define VTS  68      // v-tile LDS stride, same alignment requirement (async B128)

static __device__ __forceinline__ v8f wmma4(v2f a, v2f b, v8f c) {
  // D = A(16x4,f32) * B(4x16,f32) + C(16x16,f32)
  return __builtin_amdgcn_wmma_f32_16x16x4_f32(
      /*neg_a=*/false, a, /*neg_b=*/false, b,
      /*c_mod=*/(short)0, c, /*reuse_a=*/false, /*reuse_b=*/false);
}

// Async copy of a 64x64 f32 tile (row-major, contiguous) into LDS with row
// stride VTS, using GLOBAL_LOAD_ASYNC_TO_LDS_B128 (ASYNCcnt-tracked).
// Every lane moves 16B per issue; LDS destinations are 16B aligned.
static __device__ __forceinline__ void async_load_tile64(
    const float* __restrict__ src, float* dst, int tid, int nthreads) {
#pragma unroll 1
  for (int i = tid; i < 1024; i += nthreads) {   // 1024 chunks of 16B
    const int r = i >> 4, j = i & 15;
    const unsigned lds = (unsigned)(size_t)(&dst[r * VTS + j * 4]);
    const unsigned long long g = (unsigned long long)(size_t)(&src[r * 64 + j * 4]);
    asm volatile("global_load_async_to_lds_b128 %0, %1, off"
                 :: "v"(lds), "v"(g) : "memory");
  }
  asm volatile("s_wait_asynccnt 0x0" ::: "memory");
}

// LayerNorm over D=64 for a 16x64 tile held as 4 C/D-layout fragments.
// Element (row = r + 8*laneHi, d = t*16 + laneLo) lives in x[t][r].
static __device__ __forceinline__ void layernorm_tile(v8f x[4],
                                                      const float lw[4],
                                                      const float lb[4]) {
#pragma unroll
  for (int r = 0; r < 8; ++r) {
    float e0 = x[0][r], e1 = x[1][r], e2 = x[2][r], e3 = x[3][r];
    float s1 = e0 + e1 + e2 + e3;
    float s2 = e0 * e0 + e1 * e1 + e2 * e2 + e3 * e3;
#pragma unroll
    for (int m = 1; m < 16; m <<= 1) {   // reduce within 16-lane half-wave
      s1 += __shfl_xor(s1, m, 32);
      s2 += __shfl_xor(s2, m, 32);
    }
    const float mu  = s1 * (1.0f / 64.0f);
    const float var = s2 * (1.0f / 64.0f) - mu * mu;
    const float rs  = rsqrtf(var + 1e-5f);
    x[0][r] = (e0 - mu) * rs * lw[0] + lb[0];
    x[1][r] = (e1 - mu) * rs * lw[1] + lb[1];
    x[2][r] = (e2 - mu) * rs * lw[2] + lb[2];
    x[3][r] = (e3 - mu) * rs * lw[3] + lb[3];
  }
}

// 16x64 projection tile: out = vt[rb..rb+15][:] @ W_h^T + bias, via 64 WMMAs.
static __device__ __forceinline__ void proj_tile(
    const float* vt, int rb, int laneLo, int laneHi,
    const float* __restrict__ W, int h, const float bias[4], v8f outf[4]) {
#pragma unroll
  for (int t = 0; t < 4; ++t)
#pragma unroll
    for (int r = 0; r < 8; ++r) outf[t][r] = bias[t];
#pragma unroll
  for (int kc = 0; kc < 16; ++kc) {
    v2f a;
    const float* vp = &vt[(rb + laneLo) * VTS + kc * 4 + 2 * laneHi];
    a.x = vp[0]; a.y = vp[1];
#pragma unroll
    for (int t = 0; t < 4; ++t) {
      const float* wp = &W[(h * D_ + t * 16 + laneLo) * CIN_ + kc * 4 + 2 * laneHi];
      v2f bf; bf.x = wp[0]; bf.y = wp[1];
      outf[t] = wmma4(a, bf, outf[t]);
    }
  }
}

// Stash a 16x64 tile (C/D fragment layout) to LDS as [row][d].
static __device__ __forceinline__ void stash_tile(float* buf, int laneLo,
                                                  int laneHi, const v8f x[4]) {
#pragma unroll
  for (int t = 0; t < 4; ++t)
#pragma unroll
    for (int r = 0; r < 8; ++r)
      buf[(r + 8 * laneHi) * XS + t * 16 + laneLo] = x[t][r];
}

// ---------------------------------------------------------------------------
// Kernel 1: partial KV = LN(K)^T V over a 512-row chunk.
// Grid: 128 blocks = (b, chunk 0..15, headGroup 0..1), 256 threads = 8 waves.
// Wave pair per head: role 0 projects K (+LN), role 1 projects V; each wave
// accumulates half (eT pair) of the 64x64 KV -> only 8 acc fragments/wave.
// launch_bounds(.,4): >=4 waves/SIMD -> VGPRs capped at 256 (no MSB movs).
// ---------------------------------------------------------------------------
__global__ __launch_bounds__(256, 4) void kv_partial_kernel(
    const float* __restrict__ v,
    const float* __restrict__ Wk, const float* __restrict__ bk,
    const float* __restrict__ Wv, const float* __restrict__ bv,
    const float* __restrict__ lnkw, const float* __restrict__ lnkb,
    float* __restrict__ partial)  // [B*H][16][64*64]
{
  __shared__ float vt[64 * VTS];        // 64-row slab of v
  __shared__ float kbuf[4][16 * XS];    // per local-head normalized K tile
  __shared__ float vbuf[4][16 * XS];    // per local-head V tile

  const int tid    = threadIdx.x;
  const int wave   = tid >> 5;
  const int lane   = tid & 31;
  const int laneLo = lane & 15;
  const int laneHi = lane >> 4;
  const int hLoc   = wave >> 1;         // local head 0..3
  const int role   = wave & 1;          // 0: K producer, 1: V producer

  const int bi      = blockIdx.x;
  const int b       = bi >> 5;
  const int chunk   = (bi >> 1) & 15;
  const int hg      = bi & 1;
  const int h       = hg * 4 + hLoc;
  const int rowBase = chunk * 512;

  const float* W  = role ? Wv : Wk;
  const float* bs = role ? bv : bk;

  // warm weights into L2 (global_prefetch_b8)
  __builtin_prefetch(&W[(h * D_) * CIN_ + lane * 4], 0, 1);

  float lw[4], lb[4], pb[4];
#pragma unroll
  for (int t = 0; t < 4; ++t) {
    const int d = h * D_ + t * 16 + laneLo;
    lw[t] = lnkw[d]; lb[t] = lnkb[d];
    pb[t] = bs[d];
  }

  v8f acc[4][2];                        // KV accumulator [dTile][eT-local]
#pragma unroll
  for (int i = 0; i < 4; ++i)
#pragma unroll
    for (int j = 0; j < 2; ++j)
#pragma unroll
      for (int r = 0; r < 8; ++r) acc[i][j][r] = 0.0f;

#pragma unroll 1
  for (int stage = 0; stage < 8; ++stage) {
    __syncthreads();                    // previous stage fully consumed
    const size_t gbase = ((size_t)b * N_ + rowBase + stage * 64) * CIN_;
    async_load_tile64(v + gbase, vt, tid, 256);
    __syncthreads();                    // all waves' async copies complete

#pragma unroll 1
    for (int rt = 0; rt < 4; ++rt) {
      const int rb = rt * 16;

      // ---- projection phase: role 0 -> K (+LN), role 1 -> V ----
      v8f pf[4];
      proj_tile(vt, rb, laneLo, laneHi, W, h, pb, pf);
      if (role == 0) {
        layernorm_tile(pf, lw, lb);
        stash_tile(kbuf[hLoc], laneLo, laneHi, pf);
      } else {
        stash_tile(vbuf[hLoc], laneLo, laneHi, pf);
      }
      __syncthreads();

      // ---- KV += K^T (16-chunk of n) @ V, this wave's eT half ----
      const float* kb_ = kbuf[hLoc];
      const float* vb_ = vbuf[hLoc];
#pragma unroll
      for (int nc = 0; nc < 4; ++nc) {
        v2f aK[4];
#pragma unroll
        for (int dT = 0; dT < 4; ++dT) {
          aK[dT].x = kb_[(nc * 4 + 2 * laneHi + 0) * XS + dT * 16 + laneLo];
          aK[dT].y = kb_[(nc * 4 + 2 * laneHi + 1) * XS + dT * 16 + laneLo];
        }
#pragma unroll
        for (int eL = 0; eL < 2; ++eL) {
          const int eT = role * 2 + eL;
          v2f bf;
          bf.x = vb_[(nc * 4 + 2 * laneHi + 0) * XS + eT * 16 + laneLo];
          bf.y = vb_[(nc * 4 + 2 * laneHi + 1) * XS + eT * 16 + laneLo];
#pragma unroll
          for (int dT = 0; dT < 4; ++dT)
            acc[dT][eL] = wmma4(aK[dT], bf, acc[dT][eL]);
        }
      }
      __syncthreads();                  // tiles consumed before next overwrite
    }
  }

  // write this chunk's partial (deterministic fixed-order reduction later)
  float* pp = partial + (((size_t)(b * H_ + h) * 16 + chunk) * 4096);
#pragma unroll
  for (int dT = 0; dT < 4; ++dT)
#pragma unroll
    for (int eL = 0; eL < 2; ++eL)
#pragma unroll
      for (int r = 0; r < 8; ++r) {
        const int d = dT * 16 + r + 8 * laneHi;
        const int e = (role * 2 + eL) * 16 + laneLo;
        pp[d * 64 + e] = acc[dT][eL][r];
      }
}

// ---------------------------------------------------------------------------
// Kernel 2: fixed-order reduction of the 16 partials per (b,h).
// ---------------------------------------------------------------------------
__global__ __launch_bounds__(256, 4) void kv_reduce_kernel(
    const float* __restrict__ partial, float* __restrict__ KV)
{
  const int bh = blockIdx.x;   // 32
  const float* pp = partial + (size_t)bh * 16 * 4096;
  for (int i = threadIdx.x; i < 4096; i += 256) {
    float s = 0.0f;
#pragma unroll
    for (int c = 0; c < 16; ++c) s += pp[c * 4096 + i];
    KV[(size_t)bh * 4096 + i] = s;
  }
}

// ---------------------------------------------------------------------------
// Kernel 3: M[b,h] = KV[b,h] @ o_w_h^T   (64x64x64). 32 blocks, 4 waves.
// ---------------------------------------------------------------------------
__global__ __launch_bounds__(128, 4) void mproj_kernel(
    const float* __restrict__ KV, const float* __restrict__ ow,
    float* __restrict__ Mws)
{
  const int bh = blockIdx.x;
  const int h  = bh & 7;
  const int tid = threadIdx.x;
  const int dT  = tid >> 5;            // wave = dTile
  const int lane = tid & 31;
  const int laneLo = lane & 15;
  const int laneHi = lane >> 4;
  const float* kvp = KV + (size_t)bh * 4096;
  float* mp = Mws + (size_t)bh * 4096;

#pragma unroll
  for (int fT = 0; fT < 4; ++fT) {
    v8f acc;
#pragma unroll
    for (int r = 0; r < 8; ++r) acc[r] = 0.0f;
#pragma unroll
    for (int kc = 0; kc < 16; ++kc) {
      v2f a, bf;
      const float* ap = &kvp[(dT * 16 + laneLo) * 64 + kc * 4 + 2 * laneHi];
      a.x = ap[0]; a.y = ap[1];
      const float* bp = &ow[(fT * 16 + laneLo) * (H_ * D_) + h * D_ + kc * 4 + 2 * laneHi];
      bf.x = bp[0]; bf.y = bp[1];
      acc = wmma4(a, bf, acc);
    }
#pragma unroll
    for (int r = 0; r < 8; ++r)
      mp[(dT * 16 + r + 8 * laneHi) * 64 + fT * 16 + laneLo] = acc[r];
  }
}

// ---------------------------------------------------------------------------
// Kernel 4: out = (sum_h LN(Q_h) @ M_h + o_b) / N.  Q recomputed on the fly.
// Grid: 512 blocks x 128 threads (4 waves, wave = 16-row tile of 64).
// ---------------------------------------------------------------------------
__global__ __launch_bounds__(128, 4) void out_kernel(
    const float* __restrict__ v,
    const float* __restrict__ Wq, const float* __restrict__ bq,
    const float* __restrict__ lnqw, const float* __restrict__ lnqb,
    const float* __restrict__ Mws,
    const float* __restrict__ ob,
    float* __restrict__ out)
{
  __shared__ float vt[64 * VTS];
  __shared__ float xbuf[4][16 * XS];

  const int tid    = threadIdx.x;
  const int wave   = tid >> 5;
  const int lane   = tid & 31;
  const int laneLo = lane & 15;
  const int laneHi = lane >> 4;

  const int blocksPerB = N_ / 64;            // 128
  const int b       = blockIdx.x / blocksPerB;
  const int rowBase = (blockIdx.x % blocksPerB) * 64;
  const int rb      = wave * 16;

  __builtin_prefetch(&Wq[tid * 16], 0, 1);   // global_prefetch_b8

  const size_t gbase = ((size_t)b * N_ + rowBase) * CIN_;
  async_load_tile64(v + gbase, vt, tid, 128);
  __syncthreads();

  float obv[4];
#pragma unroll
  for (int t = 0; t < 4; ++t) obv[t] = ob[t * 16 + laneLo];

  float* xb = xbuf[wave];

  v8f oacc[4];                               // output accumulator over heads
#pragma unroll
  for (int t = 0; t < 4; ++t)
#pragma unroll
    for (int r = 0; r < 8; ++r) oacc[t][r] = 0.0f;

#pragma unroll 1
  for (int h = 0; h < H_; ++h) {
    float lw[4], lb[4], qb[4];
#pragma unroll
    for (int t = 0; t < 4; ++t) {
      const int d = h * D_ + t * 16 + laneLo;
      lw[t] = lnqw[d]; lb[t] = lnqb[d]; qb[t] = bq[d];
    }

    // ---- Q projection for this head, then LN ----
    v8f qacc[4];
    proj_tile(vt, rb, laneLo, laneHi, Wq, h, qb, qacc);
    layernorm_tile(qacc, lw, lb);

    // stash Q as [row][d] to re-read in A-fragment layout (per-wave buffer)
    stash_tile(xb, laneLo, laneHi, qacc);

    // ---- oacc += Q (16x64) @ M_h (64x64) ----
    const float* Mp = Mws + (size_t)(b * H_ + h) * 4096;
#pragma unroll
    for (int kc = 0; kc < 16; ++kc) {
      v2f a;
      a.x = xb[laneLo * XS + kc * 4 + 2 * laneHi + 0];
      a.y = xb[laneLo * XS + kc * 4 + 2 * laneHi + 1];
#pragma unroll
      for (int fT = 0; fT < 4; ++fT) {
        v2f bf;
        bf.x = Mp[(kc * 4 + 2 * laneHi + 0) * 64 + fT * 16 + laneLo];
        bf.y = Mp[(kc * 4 + 2 * laneHi + 1) * 64 + fT * 16 + laneLo];
        oacc[fT] = wmma4(a, bf, oacc[fT]);
      }
    }
  }

  const float invN = 1.0f / (float)N_;
#pragma unroll
  for (int fT = 0; fT < 4; ++fT)
#pragma unroll
    for (int r = 0; r < 8; ++r) {
      const int n = rowBase + rb + r + 8 * laneHi;
      out[((size_t)b * N_ + n) * 64 + fT * 16 + laneLo] =
          (oacc[fT][r] + obv[fT]) * invN;
    }
}

// ---------------------------------------------------------------------------
extern "C" void kernel_launch(void* const* d_in, const int* in_sizes, int n_in,
                              void* d_out, int out_size, void* d_ws, size_t ws_size,
                              hipStream_t stream) {
  (void)in_sizes; (void)n_in; (void)out_size; (void)ws_size;

  const float* v     = (const float*)d_in[0];
  const float* Wq_w  = (const float*)d_in[1];
  const float* Wq_b  = (const float*)d_in[2];
  const float* Wk_w  = (const float*)d_in[3];
  const float* Wk_b  = (const float*)d_in[4];
  const float* Wv_w  = (const float*)d_in[5];
  const float* Wv_b  = (const float*)d_in[6];
  const float* lnq_w = (const float*)d_in[7];
  const float* lnq_b = (const float*)d_in[8];
  const float* lnk_w = (const float*)d_in[9];
  const float* lnk_b = (const float*)d_in[10];
  const float* o_w   = (const float*)d_in[11];
  const float* o_b   = (const float*)d_in[12];
  float* out = (float*)d_out;

  // workspace: partials (8 MB) | KV (512 KB) | M (512 KB)
  float* partial = (float*)d_ws;
  float* KV = partial + (size_t)B_ * H_ * 16 * 4096;
  float* M  = KV + (size_t)B_ * H_ * 4096;

  kv_partial_kernel<<<128, 256, 0, stream>>>(v, Wk_w, Wk_b, Wv_w, Wv_b,
                                             lnk_w, lnk_b, partial);
  kv_reduce_kernel<<<32, 256, 0, stream>>>(partial, KV);
  mproj_kernel<<<32, 128, 0, stream>>>(KV, o_w, M);
  out_kernel<<<512, 128, 0, stream>>>(v, Wq_w, Wq_b, lnq_w, lnq_b, M, o_b, out);
}